// DCNv3_1D_27513560498556
// MI455X (gfx1250) — compile-verified
//
#include <hip/hip_runtime.h>
#include <hip/hip_bf16.h>

// ---------------------------------------------------------------------------
// DCNv3-1D forward for MI455X (gfx1250).
// N=8, L=4096, C=256, G=8, K=3, GC=32.  All fp32; matrix ops use
// V_WMMA_F32_16X16X4_F32 (full-precision CDNA5 matrix pipe).
// ---------------------------------------------------------------------------

#define NL     32768      // N * L
#define Cdim   256
#define Ldim   4096
#define Gdim   8
#define Ktap   3
#define GK     24         // G * Ktap

typedef float v2f __attribute__((ext_vector_type(2)));
typedef float v8f __attribute__((ext_vector_type(8)));

// ---------------------------------------------------------------------------
// Kernel 1/4: C = A(MxK) * B(KxN) + bias, fp32 WMMA.
// Block = 256 threads (8 waves), tile 128(M) x 128(N), K step 16.
// Wave grid 4x2; each wave computes a 32x64 patch = 2x4 WMMA 16x16 tiles.
// B is staged K-pair-interleaved in LDS so each B fragment is one aligned
// 8-byte LDS load into an even VGPR pair (no v_mov repacking before WMMA).
// Next K-tile is prefetched with global_prefetch_b8 while computing.
// ---------------------------------------------------------------------------
#define MB 128
#define NB 128
#define KB 16
#define BROW (NB + 16)    // Bs2 row stride/2: 288 floats ≡ 32 banks (mod 64)

__global__ __launch_bounds__(256)
void gemm_f32_wmma(const float* __restrict__ A, const float* __restrict__ B,
                   const float* __restrict__ bias, float* __restrict__ D,
                   int M, int Nn, int Kk) {
    __shared__ float As[MB][KB + 1];        // +1 pad -> conflict-free reads
    __shared__ float Bs2[KB / 2][BROW][2];  // {B[2kp][n], B[2kp+1][n]} pairs

    const int tid  = threadIdx.x;
    const int lane = tid & 31;
    const int wave = tid >> 5;

    const int nTilesN = Nn / NB;
    const int bm = (int)blockIdx.x / nTilesN;
    const int bn = (int)blockIdx.x % nTilesN;
    const int m0 = bm * MB, n0 = bn * NB;

    const int wy = wave >> 1, wx = wave & 1;   // 4 x 2 wave grid
    const int wm = wy * 32,   wn = wx * 64;

    const int hl   = lane & 15;                // lane within half-wave
    const int koff = (lane >> 4) * 2;          // K sub-offset per A/B layout

    // Per-thread staging coordinates
    const int arow = tid >> 1;                 // 0..127 (A tile row)
    const int acol = (tid & 1) * 8;            // 0 or 8
    const int bkp  = tid >> 5;                 // 0..7   (B K-pair)
    const int bnt  = (tid & 31) * 4;           // 0..124 (B col group)

    v8f acc[2][4] = {};

    for (int k0 = 0; k0 < Kk; k0 += KB) {
        // --- stage A tile (128x16): two float4 per thread ---
        {
            const float* gp = A + (size_t)(m0 + arow) * Kk + k0 + acol;
            const float4 v0 = *(const float4*)gp;
            const float4 v1 = *(const float4*)(gp + 4);
            float* sp = &As[arow][acol];
            sp[0] = v0.x; sp[1] = v0.y; sp[2] = v0.z; sp[3] = v0.w;
            sp[4] = v1.x; sp[5] = v1.y; sp[6] = v1.z; sp[7] = v1.w;
        }
        // --- stage B tile (16x128) K-pair-interleaved: rows 2kp and 2kp+1,
        //     4 columns each; destination is 32 contiguous bytes ---
        {
            const float* gp0 = B + (size_t)(k0 + 2 * bkp) * Nn + n0 + bnt;
            const float* gp1 = gp0 + Nn;
            const float4 v0 = *(const float4*)gp0;   // k = 2kp
            const float4 v1 = *(const float4*)gp1;   // k = 2kp+1
            float4* sp = (float4*)&Bs2[bkp][bnt][0];
            sp[0] = make_float4(v0.x, v1.x, v0.y, v1.y);
            sp[1] = make_float4(v0.z, v1.z, v0.w, v1.w);
        }
        // --- prefetch next K tile into cache (global_prefetch_b8) ---
        if (k0 + KB < Kk) {
            __builtin_prefetch(A + (size_t)(m0 + arow) * Kk + k0 + KB + acol, 0, 3);
            __builtin_prefetch(B + (size_t)(k0 + KB + 2 * bkp) * Nn + n0 + bnt, 0, 3);
        }
        __syncthreads();

        #pragma unroll
        for (int kk = 0; kk < KB; kk += 4) {
            const int kp = (kk + koff) >> 1;   // K-pair index for this lane
            v2f a[2], b[4];
            #pragma unroll
            for (int r = 0; r < 2; ++r) {
                const int m = wm + r * 16 + hl;
                a[r][0] = As[m][kk + koff];
                a[r][1] = As[m][kk + koff + 1];
            }
            #pragma unroll
            for (int c = 0; c < 4; ++c)
                b[c] = *(const v2f*)&Bs2[kp][wn + c * 16 + hl][0];
            #pragma unroll
            for (int r = 0; r < 2; ++r)
                #pragma unroll
                for (int c = 0; c < 4; ++c)
                    acc[r][c] = __builtin_amdgcn_wmma_f32_16x16x4_f32(
                        false, a[r], false, b[c], (short)0, acc[r][c],
                        false, false);
        }
        __syncthreads();
    }

    // --- write back with bias.  D layout: VGPR v -> row v (lanes 0-15)
    //     or row v+8 (lanes 16-31), col = lane%16. ---
    const int mhalf = (lane >> 4) * 8;
    #pragma unroll
    for (int c = 0; c < 4; ++c) {
        const int gn = n0 + wn + c * 16 + hl;
        const float bv = bias[gn];
        #pragma unroll
        for (int r = 0; r < 2; ++r) {
            #pragma unroll
            for (int v = 0; v < 8; ++v) {
                const int gm = m0 + wm + r * 16 + mhalf + v;
                D[(size_t)gm * Nn + gn] = acc[r][c][v] + bv;
            }
        }
    }
}

// ---------------------------------------------------------------------------
// Kernel 2: depthwise conv(K=3) + LayerNorm(C) + exact GELU + offset/mask
// linears + softmax(K).  One wave per token; lane holds channels c = lane+32j.
// ---------------------------------------------------------------------------
__global__ __launch_bounds__(256)
void feat_kernel(const float* __restrict__ x,
                 const float* __restrict__ conv_w, const float* __restrict__ conv_b,
                 const float* __restrict__ ln_g,   const float* __restrict__ ln_b,
                 const float* __restrict__ W_off,  const float* __restrict__ b_off,
                 const float* __restrict__ W_mask, const float* __restrict__ b_mask,
                 float* __restrict__ offs, float* __restrict__ masks) {
    __shared__ float sWoff[Cdim * GK];    // 24 KB
    __shared__ float sWmask[Cdim * GK];   // 24 KB
    __shared__ float sFeat[8][Cdim];      //  8 KB
    __shared__ float sOut[8][48];

    const int tid  = threadIdx.x;
    const int lane = tid & 31;
    const int wave = tid >> 5;

    for (int i = tid; i < Cdim * GK; i += 256) {
        sWoff[i]  = W_off[i];
        sWmask[i] = W_mask[i];
    }
    __syncthreads();

    const int tokBase = (int)blockIdx.x * 64 + wave * 8;

    for (int it = 0; it < 8; ++it) {
        const int tok = tokBase + it;
        const int l   = tok & (Ldim - 1);
        const size_t base = (size_t)tok * Cdim;

        // depthwise conv along L (zero pad), then running sums for LN
        float yv[8];
        float s1 = 0.f, s2 = 0.f;
        #pragma unroll
        for (int j = 0; j < 8; ++j) {
            const int c = lane + 32 * j;
            const float xm = (l > 0)        ? x[base - Cdim + c] : 0.f;
            const float x0 = x[base + c];
            const float xp = (l < Ldim - 1) ? x[base + Cdim + c] : 0.f;
            const float y  = conv_w[c * 3 + 0] * xm + conv_w[c * 3 + 1] * x0 +
                             conv_w[c * 3 + 2] * xp + conv_b[c];
            yv[j] = y; s1 += y; s2 += y * y;
        }
        #pragma unroll
        for (int o = 16; o > 0; o >>= 1) {  // wave32 reduction
            s1 += __shfl_xor(s1, o, 32);
            s2 += __shfl_xor(s2, o, 32);
        }
        const float mu   = s1 * (1.f / 256.f);
        const float var  = s2 * (1.f / 256.f) - mu * mu;
        const float rstd = rsqrtf(var + 1e-6f);

        #pragma unroll
        for (int j = 0; j < 8; ++j) {
            const int c = lane + 32 * j;
            const float yn = (yv[j] - mu) * rstd * ln_g[c] + ln_b[c];
            // exact GELU (erf form, matches approximate=False)
            sFeat[wave][c] = 0.5f * yn * (1.f + erff(yn * 0.70710678118654752f));
        }
        __syncthreads();

        // 48 outputs: o<24 -> offset (W_off), o>=24 -> mask logit (W_mask).
        // lane handles o=lane; lanes 0..15 also handle o=lane+32.
        const int o0 = lane;
        const int o1 = lane + 32;
        const float* w0p = (o0 < GK) ? &sWoff[o0] : &sWmask[o0 - GK];
        const float* w1p = &sWmask[o1 - GK];
        float r0 = 0.f, r1 = 0.f;
        #pragma unroll 8
        for (int c = 0; c < Cdim; ++c) {
            const float f = sFeat[wave][c];
            r0 += f * w0p[c * GK];
            if (lane < 16) r1 += f * w1p[c * GK];
        }
        r0 += (o0 < GK) ? b_off[o0] : b_mask[o0 - GK];
        sOut[wave][o0] = r0;
        if (lane < 16) sOut[wave][o1] = r1 + b_mask[o1 - GK];
        __syncthreads();

        if (lane < GK) offs[(size_t)tok * GK + lane] = sOut[wave][lane];
        if (lane < Gdim) {
            const float m0 = sOut[wave][GK + lane * 3 + 0];
            const float m1 = sOut[wave][GK + lane * 3 + 1];
            const float m2 = sOut[wave][GK + lane * 3 + 2];
            const float mx = fmaxf(m0, fmaxf(m1, m2));
            const float e0 = __expf(m0 - mx);
            const float e1 = __expf(m1 - mx);
            const float e2 = __expf(m2 - mx);
            const float inv = 1.f / (e0 + e1 + e2);
            masks[(size_t)tok * GK + lane * 3 + 0] = e0 * inv;
            masks[(size_t)tok * GK + lane * 3 + 1] = e1 * inv;
            masks[(size_t)tok * GK + lane * 3 + 2] = e2 * inv;
        }
        __syncthreads();
    }
}

// ---------------------------------------------------------------------------
// Kernel 3: deformable sampling.  One thread per (token, channel).
// remainder into [0,L), bilinear with zero pad at the right edge.
// ---------------------------------------------------------------------------
__global__ __launch_bounds__(256)
void sample_kernel(const float* __restrict__ xproj,
                   const float* __restrict__ offs,
                   const float* __restrict__ masks,
                   float* __restrict__ outmid) {
    const size_t id = (size_t)blockIdx.x * 256 + threadIdx.x;  // over NL*C
    const int c     = (int)(id & (Cdim - 1));
    const int tok   = (int)(id >> 8);
    const int l     = tok & (Ldim - 1);
    const int nbase = tok & ~(Ldim - 1);      // n * L
    const int g     = c >> 5;

    const float* op = offs  + (size_t)tok * GK + g * 3;
    const float* mp = masks + (size_t)tok * GK + g * 3;

    float acc = 0.f;
    #pragma unroll
    for (int k = 0; k < Ktap; ++k) {
        const float loc = (float)l + (float)(k - 1) + op[k];
        float r = fmodf(loc, (float)Ldim);
        if (r < 0.f) r += (float)Ldim;        // jnp.remainder semantics
        const float f0 = floorf(r);
        const float w1 = r - f0;
        const float w0 = 1.f - w1;
        int i0 = (int)f0;
        i0 = i0 < 0 ? 0 : (i0 > Ldim - 1 ? Ldim - 1 : i0);
        const int i1 = i0 + 1;
        const float v0 = xproj[((size_t)(nbase + i0) << 8) + c];
        const float v1 = (i1 < Ldim) ? xproj[((size_t)(nbase + i1) << 8) + c]
                                     : 0.f;
        acc += mp[k] * (w0 * v0 + w1 * v1);
    }
    outmid[id] = acc;
}

// ---------------------------------------------------------------------------
extern "C" void kernel_launch(void* const* d_in, const int* in_sizes, int n_in,
                              void* d_out, int out_size, void* d_ws, size_t ws_size,
                              hipStream_t stream) {
    (void)in_sizes; (void)n_in; (void)out_size; (void)ws_size;

    const float* x      = (const float*)d_in[0];
    const float* W_in   = (const float*)d_in[1];
    const float* b_in   = (const float*)d_in[2];
    const float* conv_w = (const float*)d_in[3];
    const float* conv_b = (const float*)d_in[4];
    const float* ln_g   = (const float*)d_in[5];
    const float* ln_b   = (const float*)d_in[6];
    const float* W_off  = (const float*)d_in[7];
    const float* b_off  = (const float*)d_in[8];
    const float* W_mask = (const float*)d_in[9];
    const float* b_mask = (const float*)d_in[10];
    const float* W_out  = (const float*)d_in[11];
    const float* b_out  = (const float*)d_in[12];
    float* out = (float*)d_out;

    float* ws     = (float*)d_ws;
    float* xproj  = ws;                                  // NL*C
    float* offs   = xproj  + (size_t)NL * Cdim;          // NL*24
    float* masksb = offs   + (size_t)NL * GK;            // NL*24
    float* outmid = masksb + (size_t)NL * GK;            // NL*C

    const dim3 blk(256);
    const dim3 gemmGrid((NL / MB) * (Cdim / NB));        // 256 * 2

    gemm_f32_wmma<<<gemmGrid, blk, 0, stream>>>(x, W_in, b_in, xproj,
                                                NL, Cdim, Cdim);
    feat_kernel<<<dim3(NL / 64), blk, 0, stream>>>(x, conv_w, conv_b, ln_g, ln_b,
                                                   W_off, b_off, W_mask, b_mask,
                                                   offs, masksb);
    sample_kernel<<<dim3((NL * Cdim) / 256), blk, 0, stream>>>(xproj, offs,
                                                               masksb, outmid);
    gemm_f32_wmma<<<gemmGrid, blk, 0, stream>>>(outmid, W_out, b_out, out,
                                                NL, Cdim, Cdim);
}